// CubeSplitter3D_8813272892075
// MI455X (gfx1250) — compile-verified
//
#include <hip/hip_runtime.h>

typedef __attribute__((ext_vector_type(2))) float v2f;
typedef __attribute__((ext_vector_type(8))) float v8f;

// Problem constants (from reference): x [2,4,192,192,192] f32, proj [512,96] f32
#define B_DIM     2
#define C_DIM     4
#define S_DIM     192
#define CSZ       8
#define NB        24            // 192/8 blocks per axis
#define KTOT      512           // 8^3
#define NE        96            // embed dim
#define NCUBES    55296         // C_DIM*NB^3
#define MTOT      (B_DIM * NCUBES)   // 110592 rows
#define KCHUNK    128
#define LDS_STRIDE (KCHUNK + 2)  // pad to spread LDS banks
#define NTILES    (NE / 16)      // 6
#define WAVES_PER_BLOCK 8

__global__ __launch_bounds__(256)
void cube_proj_wmma_f32(const float* __restrict__ x,
                        const float* __restrict__ proj,
                        float* __restrict__ out)
{
    // proj chunk, transposed: ldsB[n][kk]
    __shared__ float ldsB[NE * LDS_STRIDE];

    const int tid   = threadIdx.x;
    const int lane  = tid & 31;
    const int wave  = tid >> 5;
    const int half  = lane >> 4;   // 0: lanes 0-15, 1: lanes 16-31
    const int mlane = lane & 15;

    const int tile = blockIdx.x * WAVES_PER_BLOCK + wave;  // 16-row M tile
    const int m    = tile * 16 + mlane;                    // this lane's A row

    // decompose row -> (b, c, zb, yb, xb)  (row-major over c, zb, yb, xb)
    const int b  = m / NCUBES;
    int rem      = m % NCUBES;
    const int c  = rem / (NB * NB * NB);  rem %= (NB * NB * NB);
    const int zb = rem / (NB * NB);       rem %= (NB * NB);
    const int yb = rem / NB;
    const int xb = rem % NB;

    // element offset of this cube's origin in x
    const long cubeBase =
        ((((long)(b * C_DIM + c) * S_DIM + zb * CSZ) * S_DIM + yb * CSZ) * S_DIM
         + xb * CSZ);

    v8f acc[NTILES] = {};   // 48 accumulator VGPRs

    for (int kc = 0; kc < KTOT; kc += KCHUNK) {
        __syncthreads();  // previous chunk fully consumed
        // cooperative transpose-load of proj[kc:kc+KCHUNK, :] into LDS
        for (int idx = tid; idx < NE * KCHUNK; idx += 256) {
            const int n  = idx % NE;   // consecutive lanes -> coalesced global
            const int kk = idx / NE;
            ldsB[n * LDS_STRIDE + kk] = proj[(long)(kc + kk) * NE + n];
        }
        __syncthreads();

        for (int k0 = 0; k0 < KCHUNK; k0 += 4) {
            // A fragment: lanes 0-15 hold K=k,k+1 ; lanes 16-31 hold K=k+2,k+3
            const int k  = kc + k0 + half * 2;
            const int dz = k >> 6;
            const int dy = (k >> 3) & 7;
            const int dx = k & 7;              // even -> 8B-aligned float2
            const v2f a = *(const v2f*)(x + cubeBase
                                        + ((long)dz * S_DIM + dy) * S_DIM + dx);

#pragma unroll
            for (int nt = 0; nt < NTILES; ++nt) {
                // B fragment (4x16): lane holds [k0+2*half .. +1] of column nt*16+mlane
                const v2f bf = *(const v2f*)(
                    &ldsB[(nt * 16 + mlane) * LDS_STRIDE + k0 + half * 2]);
                acc[nt] = __builtin_amdgcn_wmma_f32_16x16x4_f32(
                    /*neg_a=*/false, a, /*neg_b=*/false, bf,
                    /*c_mod=*/(short)0, acc[nt],
                    /*reuse_a=*/false, /*reuse_b=*/false);
            }
        }
    }

    // C/D layout: VGPR r -> row (half*8 + r), column mlane within N-tile
    const int rowBase = tile * 16 + half * 8;
#pragma unroll
    for (int nt = 0; nt < NTILES; ++nt) {
#pragma unroll
        for (int r = 0; r < 8; ++r) {
            out[(long)(rowBase + r) * NE + nt * 16 + mlane] = acc[nt][r];
        }
    }
}

extern "C" void kernel_launch(void* const* d_in, const int* in_sizes, int n_in,
                              void* d_out, int out_size, void* d_ws, size_t ws_size,
                              hipStream_t stream) {
    (void)in_sizes; (void)n_in; (void)d_ws; (void)ws_size; (void)out_size;
    const float* x    = (const float*)d_in[0];
    const float* proj = (const float*)d_in[1];
    float*       out  = (float*)d_out;

    const int tiles  = MTOT / 16;                 // 6912
    const int blocks = tiles / WAVES_PER_BLOCK;   // 864
    cube_proj_wmma_f32<<<blocks, 256, 0, stream>>>(x, proj, out);
}